// PositionalEncoder_71597104824737
// MI455X (gfx1250) — compile-verified
//
#include <hip/hip_runtime.h>

// PositionalEncoder reduction: out[k] = sum_j (a[j] - (k/D)*b[j]) * E[j,k]
//   a[j] = 1 - j/J, b[j] = 1 - 2j/J  (j,k are 1-based in the reference)
// Memory-bound: 134 MB single pass over E -> ~5.8us floor @ 23.3 TB/s.
// Implemented as a 2xJ @ JxD mat-vec via V_WMMA_F32_16X16X4_F32 (weights in A,
// E tiles in B), two-stage deterministic reduction through d_ws.

#define PE_J 16384
#define PE_D 2048
#define ROW_BLOCKS 32                 // partials rows in d_ws (256 KB)
#define ROWS_PER_WAVE (PE_J / ROW_BLOCKS)   // 512
#define COL_TILES (PE_D / 16)         // 128
#define WAVES_PER_BLK 8               // 256 threads, wave32
#define S1_BLOCKS ((COL_TILES * ROW_BLOCKS) / WAVES_PER_BLK)  // 512

typedef __attribute__((ext_vector_type(2))) float v2f;
typedef __attribute__((ext_vector_type(8))) float v8f;

__global__ __launch_bounds__(256) void pe_stage1(const float* __restrict__ E,
                                                 float* __restrict__ partial) {
    const int lane  = threadIdx.x & 31;
    const int gwave = blockIdx.x * WAVES_PER_BLK + (threadIdx.x >> 5);
    const int tile  = gwave & (COL_TILES - 1);   // column tile: 16 cols
    const int rblk  = gwave >> 7;                // row block: 512 rows
    const int c0    = tile * 16;
    const int r0    = rblk * ROWS_PER_WAVE;

    const int halfsel = lane >> 4;               // 0: K=0,1  1: K=2,3
    const int m       = lane & 15;               // A-matrix row / output column

    // A-matrix weight generator: row m==0 -> a[j], m==1 -> b[j], else 0.
    const float invJ = 1.0f / (float)PE_J;
    const float base = (m <= 1) ? 1.0f : 0.0f;
    const float coef = (m == 0) ? invJ : ((m == 1) ? 2.0f * invJ : 0.0f);

    // Per-lane stream pointer: rows r0 + 2*halfsel + {0, D}, cols c0+m.
    const float* p = E + (size_t)(r0 + 2 * halfsel) * PE_D + (c0 + m);
    // 1-based j for the first row this lane touches.
    float jf = (float)(r0 + 2 * halfsel + 1);

    v8f acc = {0.f, 0.f, 0.f, 0.f, 0.f, 0.f, 0.f, 0.f};

    for (int it = 0; it < ROWS_PER_WAVE / 16; ++it) {
#pragma unroll
        for (int q = 0; q < 4; ++q) {
            // B tile: B[K=2*halfsel+v][N=m] = E[row + v*D] at immediate offsets.
            v2f b;
            b.x = p[(size_t)(4 * q) * PE_D];
            b.y = p[(size_t)(4 * q + 1) * PE_D];
            // A tile: weights for the two K-rows this lane owns.
            float j0 = jf + (float)(4 * q);
            v2f a;
            a.x = base - coef * j0;
            a.y = base - coef * (j0 + 1.0f);
            acc = __builtin_amdgcn_wmma_f32_16x16x4_f32(
                /*neg_a=*/false, a, /*neg_b=*/false, b,
                /*c_mod=*/(short)0, acc, /*reuse_a=*/false, /*reuse_b=*/false);
        }
        p  += (size_t)16 * PE_D;
        jf += 16.0f;
    }

    // D layout: lane<16, VGPR0 = Sa[c0+lane] (M=0), VGPR1 = Sb[c0+lane] (M=1).
    if (lane < 16) {
        float Sa = acc[0];
        float Sb = acc[1];
        int   c  = c0 + lane;
        float w  = (float)(c + 1) * (1.0f / (float)PE_D);
        partial[(size_t)rblk * PE_D + c] = Sa - w * Sb;
    }
}

__global__ __launch_bounds__(256) void pe_stage2(const float* __restrict__ partial,
                                                 float* __restrict__ out) {
    const int c = blockIdx.x * blockDim.x + threadIdx.x;
    if (c < PE_D) {
        float s = 0.0f;
#pragma unroll
        for (int r = 0; r < ROW_BLOCKS; ++r)
            s += partial[(size_t)r * PE_D + c];
        out[c] = s;
    }
}

extern "C" void kernel_launch(void* const* d_in, const int* in_sizes, int n_in,
                              void* d_out, int out_size, void* d_ws, size_t ws_size,
                              hipStream_t stream) {
    (void)in_sizes; (void)n_in; (void)out_size; (void)ws_size;
    const float* E   = (const float*)d_in[0];
    float*       out = (float*)d_out;
    float*       ws  = (float*)d_ws;   // ROW_BLOCKS * PE_D floats = 256 KB

    pe_stage1<<<S1_BLOCKS, 256, 0, stream>>>(E, ws);
    pe_stage2<<<PE_D / 256, 256, 0, stream>>>(ws, out);
}